// RelationNetwork_20890720928560
// MI455X (gfx1250) — compile-verified
//
#include <hip/hip_runtime.h>
#include <hip/hip_bf16.h>

// Relation network, algebraically factored:
//   A  = x @ W1[:, :256]^T   (per-node "i" embedding)
//   Bm = x @ W1[:, 256:]^T   (per-node "j" embedding)
//   agg[b,e] = max_{j>=1} ( Bm[b,j,e] + max_{i<j} A[b,i,e] ) + b1[e]
//   out = agg @ W2^T + b2
//
// Matrix math done with V_WMMA_F32_16X16X4_F32 (CDNA5 fp32 WMMA).

typedef __attribute__((ext_vector_type(2))) float v2f;
typedef __attribute__((ext_vector_type(8))) float v8f;

#define BATCH 32
#define NNODE 128
#define CIN   256
#define EMBED 512
#define OUTD  512

// ---------------------------------------------------------------------------
// Stage 1: per (batch, 16-wide embed tile) block.
//   8 waves; wave w computes rows [16w, 16w+16) of both the A tile and the
//   Bm tile (each 128x16) via fp32 WMMA over K=256, parks them in LDS, then
//   16 lanes run the prefix-max pair reduction and emit agg (+b1).
// ---------------------------------------------------------------------------
__global__ __launch_bounds__(256) void relnet_stage1(
    const float* __restrict__ X,    // (B, N, C) = (32,128,256)
    const float* __restrict__ W1,   // (EMBED, 2C) = (512, 512) row-major
    const float* __restrict__ b1,   // (512,)
    float* __restrict__ agg)        // (B, EMBED) = (32, 512)
{
    __shared__ float As[NNODE * 16];   // A tile, 8 KB
    __shared__ float Bs[NNODE * 16];   // Bm tile, 8 KB

    const int b    = blockIdx.x;        // batch
    const int e0   = blockIdx.y * 16;   // embed-column tile base
    const int wave = threadIdx.x >> 5;  // 0..7 -> row tile
    const int lane = threadIdx.x & 31;

    const int m  = lane & 15;           // row within 16x16 tile / B column
    const int kb = (lane >> 4) << 1;    // half-wave K offset: 0 or 2

    const float* Xrow = X + ((size_t)b * NNODE + (size_t)wave * 16 + m) * CIN;
    // B-fragment column e = e0 + (lane&15); W1 row-major so K runs along a row.
    const float* W1row = W1 + (size_t)(e0 + m) * (2 * CIN);

    for (int half = 0; half < 2; ++half) {
        const float* wp = W1row + half * CIN;
        v8f acc = {};
#pragma unroll 4
        for (int k = 0; k < CIN; k += 4) {
            // A 16x4 f32 frag: lane m holds {A[m,kb], A[m,kb+1]} (kb=0 or 2)
            v2f afrag = *(const v2f*)(Xrow + k + kb);
            // B 4x16 f32 frag: lane holds {B[kb, e], B[kb+1, e]} = W1[e, k+kb..]
            v2f bfrag = *(const v2f*)(wp + k + kb);
            acc = __builtin_amdgcn_wmma_f32_16x16x4_f32(
                false, afrag, false, bfrag, (short)0, acc, false, false);
        }
        // C/D 16x16 f32 layout: VGPR v -> row v (lanes 0-15) or v+8 (16-31),
        // column = lane & 15.
        float* dst = half ? Bs : As;
        const int n  = lane & 15;
        const int mo = (lane >> 4) << 3;
#pragma unroll
        for (int v = 0; v < 8; ++v)
            dst[(wave * 16 + mo + v) * 16 + n] = acc[v];
    }

    __syncthreads();

    // Prefix-max pair reduction: agg = max_{i<j}(A[i] + Bm[j]).
    if (threadIdx.x < 16) {
        const int e = threadIdx.x;
        float runA = As[e];                  // max A[0..j-1], start j=1
        float best = -3.402823466e+38f;
#pragma unroll 8
        for (int j = 1; j < NNODE; ++j) {
            best = fmaxf(best, runA + Bs[j * 16 + e]);
            runA = fmaxf(runA, As[j * 16 + e]);
        }
        agg[(size_t)b * EMBED + e0 + e] = best + b1[e0 + e];
    }
}

// ---------------------------------------------------------------------------
// Stage 2: out = agg @ W2^T + b2.  M=32, N=512, K=512.
// One wave per 16x16 output tile; 128 fp32 WMMA steps along K.
// ---------------------------------------------------------------------------
__global__ __launch_bounds__(32) void relnet_head(
    const float* __restrict__ agg,  // (32, 512)
    const float* __restrict__ W2,   // (512, 512) row-major
    const float* __restrict__ b2,   // (512,)
    float* __restrict__ out)        // (32, 512)
{
    const int row0 = blockIdx.x * 16;   // 0 or 16 (batch rows)
    const int c0   = blockIdx.y * 16;   // output-channel tile
    const int lane = threadIdx.x;

    const int m  = lane & 15;
    const int kb = (lane >> 4) << 1;

    const float* ap = agg + (size_t)(row0 + m) * EMBED;
    const float* wp = W2  + (size_t)(c0  + m) * EMBED;  // column o = c0+m, K along row

    v8f acc = {};
#pragma unroll 4
    for (int k = 0; k < EMBED; k += 4) {
        v2f afrag = *(const v2f*)(ap + k + kb);
        v2f bfrag = *(const v2f*)(wp + k + kb);
        acc = __builtin_amdgcn_wmma_f32_16x16x4_f32(
            false, afrag, false, bfrag, (short)0, acc, false, false);
    }

    const int n  = lane & 15;
    const int mo = (lane >> 4) << 3;
#pragma unroll
    for (int v = 0; v < 8; ++v) {
        const int r = row0 + mo + v;
        out[(size_t)r * OUTD + c0 + n] = acc[v] + b2[c0 + n];
    }
}

// ---------------------------------------------------------------------------
extern "C" void kernel_launch(void* const* d_in, const int* in_sizes, int n_in,
                              void* d_out, int out_size, void* d_ws, size_t ws_size,
                              hipStream_t stream) {
    const float* x  = (const float*)d_in[0];   // (32,128,256)
    const float* W1 = (const float*)d_in[1];   // (512,512)
    const float* b1 = (const float*)d_in[2];   // (512,)
    const float* W2 = (const float*)d_in[3];   // (512,512)
    const float* b2 = (const float*)d_in[4];   // (512,)
    float* out = (float*)d_out;                // (32,512)

    float* agg = (float*)d_ws;                 // 32*512 floats = 64 KB

    // Stage 1: (batch, embed-tile) grid, 8 waves per block.
    relnet_stage1<<<dim3(BATCH, EMBED / 16), 256, 0, stream>>>(x, W1, b1, agg);
    // Stage 2: one wave per 16x16 tile of the (32, 512) output.
    relnet_head<<<dim3(BATCH / 16, OUTD / 16), 32, 0, stream>>>(agg, W2, b2, out);
}